// sample_pseudo_gt_boxes_89309549953535
// MI455X (gfx1250) — compile-verified
//
#include <hip/hip_runtime.h>

// ---------------------------------------------------------------------------
// sample_pseudo_gt_boxes for MI455X (gfx1250, wave32)
//   B=2 images, R=1000 rois, C=81 classes (80 foreground), num_pad = C.
//   Kernel 1: one 1024-thread workgroup per (image, class):
//     - async global->LDS gather of strided class scores (ASYNCcnt path)
//     - 1024-wide LDS bitonic sort (descending score, index tie-break)
//     - fused box decode at gather time (no B*R*C*4 intermediate)
//     - pairwise suppression bit-matrix (1024x1024 bits = 128KB LDS,
//       exploits CDNA5's 320KB/WGP LDS) built in 16x16 tiles where the
//       area_i+area_j outer-sum uses V_WMMA_F32_16X16X4_F32
//     - exact greedy NMS = serial bit scan, then in-block compaction
//   Kernel 2: per-image prefix over class counts + padded gather + n.
// ---------------------------------------------------------------------------

#define RPAD 1024     // R (=1000) padded to a power of two for bitonic sort
#define MAXK 96       // kept dets stored per class (only first num_pad<=81 matter)
#define NMS_T 0.3f

#if defined(__HIP_DEVICE_COMPILE__)
# if __has_builtin(__builtin_amdgcn_wmma_f32_16x16x4_f32)
#  define USE_WMMA 1
# endif
# if __has_builtin(__builtin_amdgcn_global_load_async_to_lds_b32)
#  define USE_ASYNC 1
# endif
#endif

typedef float v2f __attribute__((ext_vector_type(2)));
typedef float v8f __attribute__((ext_vector_type(8)));

#if defined(USE_ASYNC)
typedef __attribute__((address_space(1))) int* as1_ptr;   // global int*
typedef __attribute__((address_space(3))) int* as3_ptr;   // LDS int*
#endif

__device__ __forceinline__ unsigned ballot32(bool p) {
#if defined(__HIP_DEVICE_COMPILE__) && __has_builtin(__builtin_amdgcn_ballot_w32)
  return __builtin_amdgcn_ballot_w32(p);      // wave32 ballot on gfx1250
#else
  return (unsigned)__ballot(p);
#endif
}

__global__ void __launch_bounds__(RPAD)
nms_per_class(const float* __restrict__ cls_prob,   // [B][R][C]
              const float* __restrict__ rois,       // [B][R][5]
              const float* __restrict__ bbox_pred,  // [B][R][4*C]
              const float* __restrict__ im_info,    // [B][3]
              const float* __restrict__ thr,        // [C]
              float* __restrict__ keptDets,         // [B][C-1][MAXK][5]
              int*   __restrict__ keptCount,        // [B][C-1]
              int B, int R, int C)
{
  extern __shared__ unsigned char smem[];
  float*    sKey  = (float*)smem;                 // RPAD
  int*      sIdx  = (int*)(sKey + RPAD);          // RPAD
  float*    sX1   = (float*)(sIdx + RPAD);        // RPAD
  float*    sY1   = sX1 + RPAD;                   // RPAD
  float*    sX2   = sY1 + RPAD;                   // RPAD
  float*    sY2   = sX2 + RPAD;                   // RPAD
  float*    sAr   = sY2 + RPAD;                   // RPAD
  unsigned* sSup  = (unsigned*)(sAr + RPAD);      // RPAD * 32 (bit matrix)
  unsigned* sKeep = sSup + RPAD * 32;             // 32
  unsigned* sPref = sKeep + 32;                   // 33

  const int tid = threadIdx.x;
  const int NC  = C - 1;
  const int b   = blockIdx.x / NC;
  const int cls = 1 + (blockIdx.x % NC);
  const float NEG_INF = -__builtin_inff();

  // ---- 1) stage class scores (strided gather, stride = C floats) ----------
  float raw = 0.0f;
#if defined(USE_ASYNC)
  if (tid < R) {
    const float* gp = cls_prob + ((size_t)b * R + tid) * C + cls;
    __builtin_amdgcn_global_load_async_to_lds_b32(
        (as1_ptr)gp, (as3_ptr)(sKey + tid), 0, 0);
  }
#else
  if (tid < R) raw = cls_prob[((size_t)b * R + tid) * C + cls];
#endif

  // clear suppression bit matrix while the async gather is in flight
  for (int w = tid; w < RPAD * 32; w += RPAD) sSup[w] = 0u;

#if defined(USE_ASYNC)
# if __has_builtin(__builtin_amdgcn_s_wait_asynccnt)
  __builtin_amdgcn_s_wait_asynccnt(0);
# else
  asm volatile("s_wait_asynccnt 0" ::: "memory");
# endif
  if (tid < R) raw = sKey[tid];
#endif

  const float t   = thr[cls];
  const bool  vld = (tid < R) && (raw > t);
  sKey[tid] = vld ? raw : NEG_INF;   // invalid / pad entries sink to the end
  sIdx[tid] = tid;
  __syncthreads();

  // ---- 2) bitonic sort descending (argsort(-s), index tie-break) ----------
  for (int k = 2; k <= RPAD; k <<= 1) {
    for (int j = k >> 1; j > 0; j >>= 1) {
      __syncthreads();
      const int i = tid, ixj = i ^ j;
      if (ixj > i) {
        float a = sKey[i], c = sKey[ixj];
        int   ia = sIdx[i], ic = sIdx[ixj];
        const bool dir = ((i & k) == 0);
        const bool sw  = (a < c) || (a == c && ia > ic);
        if (sw == dir) { sKey[i] = c; sKey[ixj] = a; sIdx[i] = ic; sIdx[ixj] = ia; }
      }
    }
  }
  __syncthreads();

  // ---- 3) fused decode + gather of boxes in sorted order ------------------
  {
    const int  r     = sIdx[tid];
    const bool valid = (sKey[tid] != NEG_INF);
    float bx1 = 0.f, by1 = 0.f, bx2 = 0.f, by2 = 0.f;
    if (valid) {
      const float* roi = rois + ((size_t)b * R + r) * 5;
      const float x1 = roi[1], y1 = roi[2], x2 = roi[3], y2 = roi[4];
      const float w  = x2 - x1 + 1.f, h  = y2 - y1 + 1.f;
      const float cx = x1 + 0.5f * w, cy = y1 + 0.5f * h;
      const float* dp = bbox_pred + ((size_t)b * R + r) * (size_t)(4 * C) + 4 * cls;
      const float d0 = dp[0] * 0.1f, d1 = dp[1] * 0.1f;
      const float d2 = dp[2] * 0.2f, d3 = dp[3] * 0.2f;
      const float pcx = fmaf(d0, w, cx), pcy = fmaf(d1, h, cy);
      const float pw  = expf(d2) * w,    ph  = expf(d3) * h;
      const float Wc = im_info[b * 3 + 1] - 1.f;
      const float Hc = im_info[b * 3 + 0] - 1.f;
      bx1 = fminf(fmaxf(pcx - 0.5f * pw, 0.f), Wc);
      by1 = fminf(fmaxf(pcy - 0.5f * ph, 0.f), Hc);
      bx2 = fminf(fmaxf(pcx + 0.5f * pw, 0.f), Wc);
      by2 = fminf(fmaxf(pcy + 0.5f * ph, 0.f), Hc);
    }
    sX1[tid] = bx1; sY1[tid] = by1; sX2[tid] = bx2; sY2[tid] = by2;
    sAr[tid] = fmaxf(bx2 - bx1, 0.f) * fmaxf(by2 - by1, 0.f);
    const unsigned km = ballot32(valid);       // keep init = valid[order]
    if ((tid & 31) == 0) sKeep[tid >> 5] = km;
  }
  __syncthreads();

  // ---- 4) pairwise suppression matrix, 16x16 tiles, WMMA area outer-sum ---
  {
    const int wave = tid >> 5;
    const int lane = tid & 31;
    const int ln   = lane & 15;
    const int half = lane >> 4;
    for (int tile = wave; tile < 64 * 64; tile += 32) {
      const int ti = tile >> 6;
      const int tj = tile & 63;
      if (tj < ti) continue;                   // only j > i contributes
      const int   j   = tj * 16 + ln;
      const float jx1 = sX1[j], jy1 = sY1[j], jx2 = sX2[j], jy2 = sY2[j];
      v8f S;
#if defined(USE_WMMA)
      // D = A(16x4) * B(4x16):  A = [area_i, 1, 0, 0], B = [1; area_j; 0; 0]
      //   => D[m][n] = area_i[m] + area_j[n]   (exact in f32)
      v2f A, Bv;
      A[0]  = half ? 0.f : sAr[ti * 16 + ln];
      A[1]  = half ? 0.f : 1.f;
      Bv[0] = half ? 0.f : 1.f;
      Bv[1] = half ? 0.f : sAr[j];
      v8f Cz = {};
      S = __builtin_amdgcn_wmma_f32_16x16x4_f32(false, A, false, Bv,
                                                (short)0, Cz, false, false);
#else
      const float aj = sAr[j];
#pragma unroll
      for (int v = 0; v < 8; ++v) S[v] = sAr[ti * 16 + v + half * 8] + aj;
#endif
#pragma unroll
      for (int v = 0; v < 8; ++v) {
        const int   i   = ti * 16 + v + half * 8;   // matches WMMA C/D layout
        const float xx1 = fmaxf(sX1[i], jx1), yy1 = fmaxf(sY1[i], jy1);
        const float xx2 = fminf(sX2[i], jx2), yy2 = fminf(sY2[i], jy2);
        const float inter = fmaxf(xx2 - xx1, 0.f) * fmaxf(yy2 - yy1, 0.f);
        const float iou   = inter / (S[v] - inter + 1e-9f);
        const bool  sup   = (iou > NMS_T) && (j > i);
        const unsigned bm = ballot32(sup);    // lanes 0-15: row i(v); 16-31: row i(v)+8
        if (lane == 0) {
          const int      word = tj >> 1;
          const unsigned sh   = (unsigned)((tj & 1) * 16);
          const unsigned lo   = bm & 0xFFFFu;
          const unsigned hi   = (bm >> 16) & 0xFFFFu;
          if (lo) atomicOr(&sSup[(ti * 16 + v) * 32 + word], lo << sh);
          if (hi) atomicOr(&sSup[(ti * 16 + v + 8) * 32 + word], hi << sh);
        }
      }
    }
  }
  __syncthreads();

  // ---- 5) exact greedy NMS = serial bit scan over the matrix --------------
  if (tid == 0) {
    for (int i = 0; i < RPAD; ++i) {
      if ((sKeep[i >> 5] >> (i & 31)) & 1u) {
        const unsigned* row = sSup + (size_t)i * 32;
#pragma unroll 4
        for (int w = 0; w < 32; ++w) sKeep[w] &= ~row[w];
      }
    }
  }
  __syncthreads();

  // ---- 6) compact kept dets (sorted order) into workspace -----------------
  if (tid < 32) sPref[tid] = (unsigned)__popc(sKeep[tid]);
  __syncthreads();
  if (tid == 0) {
    unsigned run = 0;
    for (int w = 0; w < 32; ++w) { const unsigned c = sPref[w]; sPref[w] = run; run += c; }
    sPref[32] = run;
  }
  __syncthreads();
  {
    const int  w   = tid >> 5, bit = tid & 31;
    const bool kb  = (sKeep[w] >> bit) & 1u;
    if (kb) {
      const unsigned below = bit ? (sKeep[w] & (0xFFFFFFFFu >> (32 - bit))) : 0u;
      const unsigned pos   = sPref[w] + (unsigned)__popc(below);
      if (pos < MAXK) {
        float* d = keptDets + (((size_t)b * NC + (cls - 1)) * MAXK + pos) * 5;
        d[0] = sX1[tid]; d[1] = sY1[tid]; d[2] = sX2[tid]; d[3] = sY2[tid];
        d[4] = (float)cls;
      }
    }
    if (tid == 0) keptCount[b * NC + (cls - 1)] = (int)sPref[32];
  }
}

// ---------------------------------------------------------------------------
__global__ void gather_out(const float* __restrict__ keptDets,
                           const int*   __restrict__ keptCount,
                           float* __restrict__ out, int B, int C)
{
  __shared__ int pref[257];
  __shared__ int nkeep;
  const int b       = blockIdx.x;
  const int NC      = C - 1;
  const int num_pad = C;
  if (threadIdx.x == 0) {
    int run = 0;
    for (int c = 0; c < NC; ++c) { pref[c] = run; run += keptCount[b * NC + c]; }
    pref[NC] = run;
    nkeep = run < num_pad ? run : num_pad;
  }
  for (int i = threadIdx.x; i < num_pad * 5; i += blockDim.x)
    out[(size_t)b * num_pad * 5 + i] = 0.f;   // reference scatters into zeros
  __syncthreads();
  for (int idx = threadIdx.x; idx < NC * MAXK; idx += blockDim.x) {
    const int c = idx / MAXK;
    const int k = idx % MAXK;
    if (k < keptCount[b * NC + c]) {
      const int pos = pref[c] + k;            // cumsum(keep)-1 position
      if (pos < num_pad) {
        const float* s = keptDets + (((size_t)b * NC + c) * MAXK + k) * 5;
        float* d = out + ((size_t)b * num_pad + pos) * 5;
        d[0] = s[0]; d[1] = s[1]; d[2] = s[2]; d[3] = s[3]; d[4] = s[4];
      }
    }
  }
  if (threadIdx.x == 0) out[(size_t)B * num_pad * 5 + b] = (float)nkeep;
}

// ---------------------------------------------------------------------------
extern "C" void kernel_launch(void* const* d_in, const int* in_sizes, int n_in,
                              void* d_out, int out_size, void* d_ws, size_t ws_size,
                              hipStream_t stream) {
  const float* cls_prob  = (const float*)d_in[0];   // [B][R][C]
  const float* rois      = (const float*)d_in[1];   // [B][R][5]
  const float* bbox_pred = (const float*)d_in[2];   // [B][R][4C]
  const float* im_info   = (const float*)d_in[3];   // [B][3]
  const float* thr       = (const float*)d_in[4];   // [C]
  (void)n_in; (void)out_size; (void)ws_size;

  const int B = in_sizes[3] / 3;                    // 2
  const int C = in_sizes[4];                        // 81
  const int R = in_sizes[0] / (B * C);              // 1000

  float* keptDets  = (float*)d_ws;                               // B*(C-1)*MAXK*5
  int*   keptCount = (int*)(keptDets + (size_t)B * (C - 1) * MAXK * 5);

  // LDS: 7 f32 arrays + 128KB bit matrix + keep/prefix words (~156KB < 320KB/WGP)
  const size_t lds = (size_t)(7 * RPAD) * 4 + (size_t)RPAD * 32 * 4 + 65 * 4;

  nms_per_class<<<B * (C - 1), RPAD, lds, stream>>>(
      cls_prob, rois, bbox_pred, im_info, thr, keptDets, keptCount, B, R, C);
  gather_out<<<B, 256, 0, stream>>>(keptDets, keptCount, (float*)d_out, B, C);
}